// GCNNet_27522150433341
// MI455X (gfx1250) — compile-verified
//
#include <hip/hip_runtime.h>
#include <hip/hip_bf16.h>
#include <math.h>

// ---------------------------------------------------------------------------
// GCN 3-layer forward for MI455X (gfx1250, wave32).
//  - GEMM via V_WMMA_F32_16X16X4_F32 (fp32-exact matrix pipe; GEMM is
//    memory-bound so lower precision gains nothing).
//  - Edge aggregation via native fp32 global atomics (relaxed, agent scope).
//  - Wave-per-node L2-normalize + ReLU with wave32 shfl reductions.
// ---------------------------------------------------------------------------

typedef float v2f __attribute__((ext_vector_type(2)));
typedef float v8f __attribute__((ext_vector_type(8)));

#define D 128          // feature dim (fixed by reference)
#define EPSN 1e-12f

__device__ __forceinline__ void atomAddF(float* p, float v) {
  __hip_atomic_fetch_add(p, v, __ATOMIC_RELAXED, __HIP_MEMORY_SCOPE_AGENT);
}

// ---- init: deg=1 (self loop), zero graph sums + counts ---------------------
__global__ void k_init(float* __restrict__ deg, float* __restrict__ gsum,
                       float* __restrict__ cnt, int N, int GS, int G) {
  int t = blockIdx.x * blockDim.x + threadIdx.x;
  if (t < N)  deg[t]  = 1.0f;
  if (t < GS) gsum[t] = 0.0f;
  if (t < G)  cnt[t]  = 0.0f;
}

// ---- degree accumulation ---------------------------------------------------
__global__ void k_deg(const int* __restrict__ dst, float* __restrict__ deg, int E) {
  int t = blockIdx.x * blockDim.x + threadIdx.x;
  if (t < E) atomAddF(&deg[dst[t]], 1.0f);
}

// ---- deg -> dinv = rsqrt(deg) (in place) -----------------------------------
__global__ void k_dinv(float* __restrict__ d, int N) {
  int t = blockIdx.x * blockDim.x + threadIdx.x;
  if (t < N) d[t] = rsqrtf(d[t]);
}

// ---- zero a float buffer (float4 stores) -----------------------------------
__global__ void k_zero4(float4* __restrict__ p, int n4) {
  int t = blockIdx.x * blockDim.x + threadIdx.x;
  if (t < n4) p[t] = make_float4(0.f, 0.f, 0.f, 0.f);
}

// ---- H = X @ W, X:[N,128], W:[128,128], via WMMA f32 16x16x4 ---------------
// Block: 256 thr = 8 waves. Block owns 16 rows of X; wave w owns cols 16w..16w+15.
// W staged in LDS (64 KB).
__global__ __launch_bounds__(256) void k_gemm_wmma(
    const float* __restrict__ X, const float* __restrict__ W,
    float* __restrict__ H, int N) {
  __shared__ float Wlds[D * D];  // 64 KB

  const int t = threadIdx.x;
  // cooperative W load: 16384 floats as 4096 float4 by 256 threads
  {
    const float4* wsrc = (const float4*)W;
    float4* wdst = (float4*)Wlds;
#pragma unroll
    for (int i = 0; i < 16; ++i) wdst[t + 256 * i] = wsrc[t + 256 * i];
  }
  __syncthreads();

  const int lane = t & 31;
  const int wave = t >> 5;
  const int mrow = lane & 15;   // M (A) or N (B/C/D) within tile
  const int half = lane >> 4;   // K-pair selector for A/B fragments
  const int col0 = wave * 16;
  const long rowBase = (long)blockIdx.x * 16;

  // A fragment source: lanes 0-15 -> K={kk,kk+1}, lanes 16-31 -> K={kk+2,kk+3}
  long arow = rowBase + mrow;
  if (arow > (long)N - 1) arow = (long)N - 1;  // keep EXEC full; stores guarded
  const float* aptr = X + arow * D + half * 2;

  v8f acc = {0.f, 0.f, 0.f, 0.f, 0.f, 0.f, 0.f, 0.f};

#pragma unroll 8
  for (int kk = 0; kk < D; kk += 4) {
    const float2 av = *(const float2*)(aptr + kk);
    v2f a; a.x = av.x; a.y = av.y;
    const int kb = kk + half * 2;
    v2f b;
    b.x = Wlds[kb * D + col0 + mrow];
    b.y = Wlds[(kb + 1) * D + col0 + mrow];
    acc = __builtin_amdgcn_wmma_f32_16x16x4_f32(
        /*neg_a=*/false, a, /*neg_b=*/false, b,
        /*c_mod=*/(short)0, acc, /*reuse_a=*/false, /*reuse_b=*/false);
  }

  // D layout: lane -> N=col0+mrow ; VGPR i -> M = half*8 + i
  const int col = col0 + mrow;
  float* hp = H + (rowBase + half * 8) * D + col;
  if (rowBase + 16 <= (long)N) {
    // full tile (always taken for N % 16 == 0): straight-line stores
#pragma unroll
    for (int i = 0; i < 8; ++i) hp[(long)i * D] = acc[i];
  } else {
#pragma unroll
    for (int i = 0; i < 8; ++i) {
      if (rowBase + half * 8 + i < (long)N) hp[(long)i * D] = acc[i];
    }
  }
}

// ---- edge aggregation: AGG[dst] += H[src] * (dinv[src]*dinv[dst]) ----------
// One wave per edge, 4 floats per lane (float4 gather, 4 scalar fp32 atomics).
__global__ __launch_bounds__(256) void k_agg(
    const float* __restrict__ H, const int* __restrict__ src,
    const int* __restrict__ dst, const float* __restrict__ dinv,
    float* __restrict__ AGG, int E) {
  const int e = blockIdx.x * 8 + (threadIdx.x >> 5);
  const int lane = threadIdx.x & 31;
  if (e >= E) return;
  const int s = src[e], d = dst[e];
  const float w = dinv[s] * dinv[d];
  const float4 hv = *(const float4*)(H + (long)s * D + lane * 4);
  float* ap = AGG + (long)d * D + lane * 4;
  atomAddF(ap + 0, hv.x * w);
  atomAddF(ap + 1, hv.y * w);
  atomAddF(ap + 2, hv.z * w);
  atomAddF(ap + 3, hv.w * w);
}

// ---- finalize: x = relu(normalize(AGG + H*self_norm + b)) ------------------
// One wave per node; lane owns cols {lane, lane+32, lane+64, lane+96}.
__global__ __launch_bounds__(256) void k_finalize(
    const float* __restrict__ AGG, const float* __restrict__ H,
    const float* __restrict__ dinv, const float* __restrict__ bias,
    float* __restrict__ OUT, int N) {
  const int n = blockIdx.x * 8 + (threadIdx.x >> 5);
  const int lane = threadIdx.x & 31;
  if (n >= N) return;
  float sn = dinv[n];
  sn *= sn;  // self-loop norm = dinv^2
  float v[4];
  float ss = 0.f;
#pragma unroll
  for (int j = 0; j < 4; ++j) {
    const int c = lane + 32 * j;
    const long idx = (long)n * D + c;
    const float val = AGG[idx] + H[idx] * sn + bias[c];
    v[j] = val;
    ss += val * val;
  }
#pragma unroll
  for (int off = 16; off > 0; off >>= 1) ss += __shfl_xor(ss, off, 32);
  const float inv = 1.0f / fmaxf(sqrtf(ss), EPSN);
#pragma unroll
  for (int j = 0; j < 4; ++j) {
    const int c = lane + 32 * j;
    OUT[(long)n * D + c] = fmaxf(v[j] * inv, 0.0f);
  }
}

// ---- mean-pool accumulation ------------------------------------------------
__global__ __launch_bounds__(256) void k_pool(
    const float* __restrict__ X, const int* __restrict__ batch,
    float* __restrict__ gsum, float* __restrict__ cnt, int N) {
  const int n = blockIdx.x * 8 + (threadIdx.x >> 5);
  const int lane = threadIdx.x & 31;
  if (n >= N) return;
  const int b = batch[n];
#pragma unroll
  for (int j = 0; j < 4; ++j) {
    const int c = lane + 32 * j;
    atomAddF(&gsum[(long)b * D + c], X[(long)n * D + c]);
  }
  if (lane == 0) atomAddF(&cnt[b], 1.0f);
}

// ---- divide by counts (in place on d_out graph region) ---------------------
__global__ void k_div(float* __restrict__ g, const float* __restrict__ cnt, int GS) {
  int t = blockIdx.x * blockDim.x + threadIdx.x;
  if (t < GS) g[t] /= fmaxf(cnt[t >> 7], 1.0f);
}

// ---------------------------------------------------------------------------
extern "C" void kernel_launch(void* const* d_in, const int* in_sizes, int n_in,
                              void* d_out, int out_size, void* d_ws, size_t ws_size,
                              hipStream_t stream) {
  const float* x     = (const float*)d_in[0];
  const int*   ei    = (const int*)d_in[1];
  const int*   batch = (const int*)d_in[2];
  const float* W[3]  = {(const float*)d_in[3], (const float*)d_in[5], (const float*)d_in[7]};
  const float* B[3]  = {(const float*)d_in[4], (const float*)d_in[6], (const float*)d_in[8]};

  const int N = in_sizes[0] / D;      // 50000
  const int E = in_sizes[1] / 2;      // 800000
  const int G = (out_size - N * D) / D;  // 500
  const int GS = G * D;

  float* out  = (float*)d_out;
  float* gemb = out;                      // [G,128]
  float* nemb = out + (size_t)GS;         // [N,128]

  float* buf0 = (float*)d_ws;             // x / agg ping buffer [N,128]
  float* buf1 = buf0 + (size_t)N * D;     // h buffer            [N,128]
  float* dinv = buf1 + (size_t)N * D;     // [N]
  float* cnt  = dinv + N;                 // [G]

  const int* src = ei;
  const int* dst = ei + E;

  const int TB = 256;
  int initn = (N > GS ? N : GS);
  k_init<<<(initn + TB - 1) / TB, TB, 0, stream>>>(dinv, gemb, cnt, N, GS, G);
  k_deg<<<(E + TB - 1) / TB, TB, 0, stream>>>(dst, dinv, E);
  k_dinv<<<(N + TB - 1) / TB, TB, 0, stream>>>(dinv, N);

  const float* xin = x;
  const int rowTiles = (N + 15) / 16;       // 3125
  const int n4 = N * (D / 4);               // float4 count for zeroing
  for (int l = 0; l < 3; ++l) {
    k_gemm_wmma<<<rowTiles, TB, 0, stream>>>(xin, W[l], buf1, N);
    k_zero4<<<(n4 + TB - 1) / TB, TB, 0, stream>>>((float4*)buf0, n4);
    k_agg<<<(E + 7) / 8, TB, 0, stream>>>(buf1, src, dst, dinv, buf0, E);
    float* xdst = (l == 2) ? nemb : buf0;
    k_finalize<<<(N + 7) / 8, TB, 0, stream>>>(buf0, buf1, dinv, B[l], xdst, N);
    xin = xdst;
  }

  k_pool<<<(N + 7) / 8, TB, 0, stream>>>(nemb, batch, gemb, cnt, N);
  k_div<<<(GS + TB - 1) / TB, TB, 0, stream>>>(gemb, cnt, GS);
}